// EncoderLayer_83245056131085
// MI455X (gfx1250) — compile-verified
//
#include <hip/hip_runtime.h>
#include <hip/hip_bf16.h>
#include <math.h>

// Problem constants (from reference)
#define B_ 64
#define L_ 200
#define D_ 64     // model dim
#define A_ 64     // attn dim (2 heads x depth 32)
#define BL (B_ * L_)          // 12800 positions
#define NEGF (-4294967296.0f) // fp32(-(2^32)+1)
#define SCALE 0.17677669529663687f // 1/sqrt(32)

typedef __attribute__((ext_vector_type(2))) float v2f;
typedef __attribute__((ext_vector_type(4))) float f32x4;
typedef __attribute__((ext_vector_type(8))) float v8f;

// ---------------------------------------------------------------------------
// Kernel 1: per-position LayerNorm (eps=1e-8, gamma/beta are [L,D]) + the
// key/query mask sums. One wave32 per position, 2 elements per lane.
// ---------------------------------------------------------------------------
__global__ void __launch_bounds__(256) ln_kernel(
    const float* __restrict__ x, const float* __restrict__ gamma,
    const float* __restrict__ beta, float* __restrict__ xn,
    float* __restrict__ keysum, float* __restrict__ qsum)
{
  const int wid  = blockIdx.x * 8 + (threadIdx.x >> 5); // position in [0,BL)
  const unsigned lane = threadIdx.x & 31;
  const int l = wid % L_;

  const float* xr = x + (size_t)wid * D_;
  float a = xr[lane], b = xr[lane + 32];

  float s = a + b;
  #pragma unroll
  for (int off = 16; off; off >>= 1) s += __shfl_xor(s, off, 32);
  float mean = s * (1.0f / 64.0f);

  float da = a - mean, db = b - mean;
  float v = da * da + db * db;
  #pragma unroll
  for (int off = 16; off; off >>= 1) v += __shfl_xor(v, off, 32);
  float inv = rsqrtf(v * (1.0f / 64.0f) + 1e-8f);

  const float* g  = gamma + (size_t)l * D_;
  const float* be = beta  + (size_t)l * D_;
  float n1 = g[lane]      * (da * inv) + be[lane];
  float n2 = g[lane + 32] * (db * inv) + be[lane + 32];
  xn[(size_t)wid * D_ + lane]      = n1;
  xn[(size_t)wid * D_ + lane + 32] = n2;

  float qs = n1 + n2;
  #pragma unroll
  for (int off = 16; off; off >>= 1) qs += __shfl_xor(qs, off, 32);
  if (lane == 0) { keysum[wid] = s; qsum[wid] = qs; }
}

// ---------------------------------------------------------------------------
// WMMA helper: D(16x16) += A(16x64) * W(64x64 col-tile n0), fp32 WMMA K=4.
// ISA 7.12.2 layouts (wave32):
//   A 16x4 : lanes 0-15 hold M=lane, VGPR0=K0/VGPR1=K1; lanes 16-31 K2/K3
//   B 4x16 : lanes 0-15 hold N=lane, VGPR0=K0/VGPR1=K1; lanes 16-31 K2/K3
//   C/D    : VGPR r -> (M=r, N=lane) for lanes 0-15, (M=r+8) for lanes 16-31
// ---------------------------------------------------------------------------
__device__ inline v8f wmma_16x16x64(const float* __restrict__ Alds, // 16x64
                                    const float* __restrict__ Wlds, // 64x64
                                    int n0, unsigned lane)
{
  v8f acc = {};
  const int m    = lane & 15;
  const int ksel = (lane >> 4) << 1; // 0 for lanes 0-15, 2 for lanes 16-31
  #pragma unroll
  for (int k0 = 0; k0 < 64; k0 += 4) {
    v2f a, b;
    a.x = Alds[m * 64 + k0 + ksel];
    a.y = Alds[m * 64 + k0 + ksel + 1];
    b.x = Wlds[(k0 + ksel)     * 64 + n0 + m];
    b.y = Wlds[(k0 + ksel + 1) * 64 + n0 + m];
    acc = __builtin_amdgcn_wmma_f32_16x16x4_f32(false, a, false, b,
                                                (short)0, acc, false, false);
  }
  return acc;
}

// ---------------------------------------------------------------------------
// Kernel 2: generic [BL,64] @ [64,64] GEMM, 16-row tile per block, 4 waves
// (each wave owns a 16-col tile). Fuses bias, +pos, relu, +residual, *rowscale.
// ---------------------------------------------------------------------------
__global__ void __launch_bounds__(128) gemm64_wmma(
    const float* __restrict__ src, const float* __restrict__ W,
    const float* __restrict__ bias, const float* __restrict__ pos,
    const float* __restrict__ resid, const float* __restrict__ rowscale,
    float* __restrict__ out, int do_relu)
{
  __shared__ float Alds[16 * 64];
  __shared__ float Wlds[64 * 64];
  const int tid  = threadIdx.x;
  const unsigned lane = tid & 31;
  const int wave = tid >> 5;           // 0..3 -> column tile
  const int row0 = blockIdx.x * 16;

  // cooperative tile loads (float4)
  #pragma unroll
  for (int i = tid; i < 16 * 16; i += 128)
    ((f32x4*)Alds)[i] = ((const f32x4*)(src + (size_t)row0 * 64))[i];
  #pragma unroll
  for (int i = tid; i < 64 * 16; i += 128)
    ((f32x4*)Wlds)[i] = ((const f32x4*)W)[i];
  __syncthreads();

  const int n0 = wave * 16;
  v8f acc = wmma_16x16x64(Alds, Wlds, n0, lane);

  const int col   = n0 + (lane & 15);
  const int rbase = row0 + ((lane >> 4) << 3); // +8 for upper-half lanes
  const float bc  = bias[col];
  #pragma unroll
  for (int r = 0; r < 8; ++r) {
    const int row = rbase + r;
    float v = acc[r] + bc;
    if (pos)      v += pos[(size_t)row * 64 + col];
    if (do_relu)  v  = fmaxf(v, 0.0f);
    if (resid)    v += resid[(size_t)row * 64 + col];
    if (rowscale) v *= rowscale[row];
    out[(size_t)row * 64 + col] = v;
  }
}

// ---------------------------------------------------------------------------
// Block reductions (256 threads)
// ---------------------------------------------------------------------------
__device__ inline float blockMax(float v, float* red, int tid) {
  red[tid] = v; __syncthreads();
  #pragma unroll
  for (int s = 128; s > 0; s >>= 1) {
    if (tid < s) red[tid] = fmaxf(red[tid], red[tid + s]);
    __syncthreads();
  }
  float r = red[0]; __syncthreads();
  return r;
}
__device__ inline float blockSum(float v, float* red, int tid) {
  red[tid] = v; __syncthreads();
  #pragma unroll
  for (int s = 128; s > 0; s >>= 1) {
    if (tid < s) red[tid] += red[tid + s];
    __syncthreads();
  }
  float r = red[0]; __syncthreads();
  return r;
}

// ---------------------------------------------------------------------------
// Kernel 3: fused attention core, one block per (b,q). Streams the causal
// (k <= q) slice of tK/tV exactly once with nontemporal loads.
// ---------------------------------------------------------------------------
__global__ void __launch_bounds__(256) attn_kernel(
    const float* __restrict__ Q, const float* __restrict__ Keff,
    const float* __restrict__ Veff, const float* __restrict__ tK,
    const float* __restrict__ tV, const float* __restrict__ xn,
    const float* __restrict__ keysum, const float* __restrict__ qsum,
    float* __restrict__ attn_out)
{
  const int bq  = blockIdx.x;        // b*L + q
  const int b   = bq / L_;
  const int q   = bq - b * L_;
  const int tid = threadIdx.x;

  __shared__ float sQ[64];
  __shared__ float w0[256], w1[256]; // per-k attention weights (2 heads)
  __shared__ float red[256];
  __shared__ float part[4][64];

  if (tid < 64) sQ[tid] = Q[(size_t)bq * 64 + tid];
  __syncthreads();

  // ---- scores: thread tid handles key k=tid; skip k>q (exact zeros) ----
  float s0 = -INFINITY, s1 = -INFINITY;
  const int k = tid;
  if (k <= q) { // (q < L_ always, so k < L_)
    const f32x4* kr  = (const f32x4*)(Keff + ((size_t)(b * L_ + k)) * 64);
    const f32x4* tkr = (const f32x4*)(tK + ((size_t)bq * L_ + k) * 64);
    const f32x4* qr  = (const f32x4*)sQ;
    float a0 = 0.0f, a1 = 0.0f;
    #pragma unroll
    for (int i = 0; i < 16; ++i) {          // 64 dims, head split at i==8
      f32x4 kv = kr[i];
      f32x4 tv = __builtin_nontemporal_load(tkr + i);
      f32x4 qv = qr[i];
      float d = qv.x * (kv.x + tv.x) + qv.y * (kv.y + tv.y) +
                qv.z * (kv.z + tv.z) + qv.w * (kv.w + tv.w);
      if (i < 8) a0 += d; else a1 += d;
    }
    const bool dead = (keysum[b * L_ + k] == 0.0f);
    s0 = dead ? NEGF : a0 * SCALE;
    s1 = dead ? NEGF : a1 * SCALE;
  }

  // ---- softmax over k (per head), then * query_mask ----
  const float m0 = blockMax(s0, red, tid);
  const float m1 = blockMax(s1, red, tid);
  const float e0 = __expf(s0 - m0);  // exp(-inf)=0 for padded/causal k
  const float e1 = __expf(s1 - m1);
  const float z0 = blockSum(e0, red, tid);
  const float z1 = blockSum(e1, red, tid);
  const float qm = (qsum[bq] != 0.0f) ? 1.0f : 0.0f;
  w0[tid] = e0 * (qm / z0);
  w1[tid] = e1 * (qm / z1);
  __syncthreads();

  // ---- out[d] = sum_{k<=q} attn[h(d)][k] * (Veff[k,d] + tV[q,k,d]) ----
  const int d = tid & 63;
  const int c = tid >> 6;                    // 4 k-chunks
  const float* attn = (d < 32) ? w0 : w1;
  const int kcount = q + 1;
  const int k0 = (c * kcount) >> 2;
  const int k1 = ((c + 1) * kcount) >> 2;
  const float* vb  = Veff + (size_t)(b * L_) * 64 + d;
  const float* tvb = tV + ((size_t)bq * L_) * 64 + d;
  float accd = 0.0f;
  for (int kk = k0; kk < k1; ++kk) {
    float vv = vb[(size_t)kk * 64] +
               __builtin_nontemporal_load(tvb + (size_t)kk * 64);
    accd = fmaf(attn[kk], vv, accd);
  }
  part[c][d] = accd;
  __syncthreads();

  if (tid < 64) { // merge partials + residual (attn_out = out + x_norm)
    float o = part[0][tid] + part[1][tid] + part[2][tid] + part[3][tid] +
              xn[(size_t)bq * 64 + tid];
    attn_out[(size_t)bq * 64 + tid] = o;
  }
}

// ---------------------------------------------------------------------------
// Launch: LN -> {Q, K+pK, V+pV} WMMA GEMMs -> fused attention -> FFN (2 WMMA
// GEMMs with relu / residual / mask fused). All on `stream`.
// ---------------------------------------------------------------------------
extern "C" void kernel_launch(void* const* d_in, const int* in_sizes, int n_in,
                              void* d_out, int out_size, void* d_ws, size_t ws_size,
                              hipStream_t stream)
{
  const float* x    = (const float*)d_in[0];
  const float* tK   = (const float*)d_in[1];
  const float* tV   = (const float*)d_in[2];
  const float* posK = (const float*)d_in[3];
  const float* posV = (const float*)d_in[4];
  const float* mask = (const float*)d_in[5];
  const float* gam  = (const float*)d_in[6];
  const float* bet  = (const float*)d_in[7];
  const float* Wq   = (const float*)d_in[8];
  const float* bqv  = (const float*)d_in[9];
  const float* Wk   = (const float*)d_in[10];
  const float* bk   = (const float*)d_in[11];
  const float* Wv   = (const float*)d_in[12];
  const float* bv   = (const float*)d_in[13];
  const float* W1   = (const float*)d_in[14];
  const float* b1   = (const float*)d_in[15];
  const float* W2   = (const float*)d_in[16];
  const float* b2   = (const float*)d_in[17];

  float* ws     = (float*)d_ws;
  float* xn     = ws;                 // BL*64
  float* Qb     = xn   + (size_t)BL * 64;
  float* Keff   = Qb   + (size_t)BL * 64;
  float* Veff   = Keff + (size_t)BL * 64;
  float* aout   = Veff + (size_t)BL * 64;
  float* keysum = aout + (size_t)BL * 64;
  float* qsum   = keysum + BL;
  float* h1     = Qb;                 // alias: Q dead after attention

  ln_kernel<<<BL / 8, 256, 0, stream>>>(x, gam, bet, xn, keysum, qsum);

  gemm64_wmma<<<BL / 16, 128, 0, stream>>>(xn, Wq, bqv, nullptr, nullptr, nullptr, Qb, 0);
  gemm64_wmma<<<BL / 16, 128, 0, stream>>>(x,  Wk, bk,  posK,    nullptr, nullptr, Keff, 0);
  gemm64_wmma<<<BL / 16, 128, 0, stream>>>(x,  Wv, bv,  posV,    nullptr, nullptr, Veff, 0);

  attn_kernel<<<BL, 256, 0, stream>>>(Qb, Keff, Veff, tK, tV, xn, keysum, qsum, aout);

  gemm64_wmma<<<BL / 16, 128, 0, stream>>>(aout, W1, b1, nullptr, nullptr, nullptr, h1, 1);
  gemm64_wmma<<<BL / 16, 128, 0, stream>>>(h1,   W2, b2, nullptr, aout,    mask, (float*)d_out, 0);
}